// KernelConv_88347477279305
// MI455X (gfx1250) — compile-verified
//
#include <hip/hip_runtime.h>

// ---------------- problem constants (from reference setup_inputs) ----------
constexpr int BS = 4;      // batch
constexpr int NB = 8;      // burst frames
constexpr int Hd = 256;    // height
constexpr int Wd = 256;    // width
constexpr int KS = 5;      // kernel size
constexpr int KK = KS * KS;
constexpr int HALO = KS / 2;

constexpr int TW = 32;     // tile width  (pixels per block in x)
constexpr int TH = 16;     // tile height (pixels per block in y)
constexpr int LW = TW + 2 * HALO;   // 36
constexpr int LH = TH + 2 * HALO;   // 20
constexpr int PRED_SZ = BS * Hd * Wd;  // pred_img element count (output 0)

#if defined(__gfx1250__) && __has_builtin(__builtin_amdgcn_global_load_async_to_lds_b32)
#define HAVE_ASYNC_LDS 1
#endif

__global__ __launch_bounds__(256) void kpn_filter_kernel(
    const float* __restrict__ frames,   // (BS, NB, H, W)
    const float* __restrict__ core,     // flat, indexed (((b*NB+n)*H+y)*W+x)*25 + t
    float* __restrict__ out)            // [pred_img | pred_img_i]
{
    __shared__ float tile[LH * LW];

    const int t   = threadIdx.x;
    const int tx  = t & (TW - 1);       // 0..31
    const int ty  = t >> 5;             // 0..7  (two rows per thread: ty, ty+8)
    const int x0  = blockIdx.x * TW;
    const int y0  = blockIdx.y * TH;
    const int b   = blockIdx.z;

    const int gx  = x0 + tx;
    const int gy0 = y0 + ty;
    const int gy1 = gy0 + 8;

    float sum0 = 0.0f, sum1 = 0.0f;

    for (int n = 0; n < NB; ++n) {
        const float* fbase = frames + (size_t)((b * NB + n) * Hd) * Wd;

        // ---- stage haloed frame tile into LDS (zero-padded OOB) ----
        for (int e = t; e < LH * LW; e += 256) {
            const int ly = e / LW;
            const int lx = e - ly * LW;
            const int fy = y0 - HALO + ly;
            const int fx = x0 - HALO + lx;
            const bool inb = (fy >= 0) && (fy < Hd) && (fx >= 0) && (fx < Wd);
#if HAVE_ASYNC_LDS
            if (inb) {
                __builtin_amdgcn_global_load_async_to_lds_b32(
                    (__attribute__((address_space(1))) int*)(fbase + fy * Wd + fx),
                    (__attribute__((address_space(3))) int*)(&tile[e]),
                    0, 0);
            } else {
                tile[e] = 0.0f;
            }
#else
            tile[e] = inb ? fbase[fy * Wd + fx] : 0.0f;
#endif
        }
#if HAVE_ASYNC_LDS
#if __has_builtin(__builtin_amdgcn_s_wait_asynccnt)
        __builtin_amdgcn_s_wait_asynccnt(0);
#else
        asm volatile("s_wait_asynccnt 0x0" ::: "memory");
#endif
#endif
        __syncthreads();

        // ---- per-pixel 25-tap dynamic filter (2 pixels per thread) ----
        const size_t pix0 = (size_t)((b * NB + n) * Hd + gy0) * Wd + gx;
        const size_t pix1 = (size_t)((b * NB + n) * Hd + gy1) * Wd + gx;
        const float* __restrict__ c0 = core + pix0 * KK;  // 25 contiguous dwords
        const float* __restrict__ c1 = core + pix1 * KK;

        float a0 = 0.0f, a1 = 0.0f;
#pragma unroll
        for (int i = 0; i < KS; ++i) {
#pragma unroll
            for (int j = 0; j < KS; ++j) {
                const int q = i * KS + j;
                a0 = fmaf(c0[q], tile[(ty + i) * LW + tx + j], a0);
                a1 = fmaf(c1[q], tile[(ty + 8 + i) * LW + tx + j], a1);
            }
        }

        out[PRED_SZ + pix0] = a0;   // pred_img_i
        out[PRED_SZ + pix1] = a1;
        sum0 += a0;
        sum1 += a1;

        __syncthreads();            // protect LDS tile before next stage
    }

    // pred_img = mean over burst
    out[(size_t)(b * Hd + gy0) * Wd + gx] = sum0 * 0.125f;
    out[(size_t)(b * Hd + gy1) * Wd + gx] = sum1 * 0.125f;
}

extern "C" void kernel_launch(void* const* d_in, const int* in_sizes, int n_in,
                              void* d_out, int out_size, void* d_ws, size_t ws_size,
                              hipStream_t stream) {
    (void)in_sizes; (void)n_in; (void)out_size; (void)d_ws; (void)ws_size;
    const float* frames = (const float*)d_in[0];
    const float* core   = (const float*)d_in[1];
    float* out          = (float*)d_out;

    dim3 grid(Wd / TW, Hd / TH, BS);   // 8 x 16 x 4 = 512 blocks
    dim3 block(256, 1, 1);             // 8 wave32s
    kpn_filter_kernel<<<grid, block, 0, stream>>>(frames, core, out);
}